// FiberBundleConnection_62826781605946
// MI455X (gfx1250) — compile-verified
//
#include <hip/hip_runtime.h>

// ---------------------------------------------------------------------------
// FiberBundleConnection transport on MI455X (gfx1250)
//
// v_b <- v_b @ M_{idx[b,t]} for 511 steps. idx depends only on base_path, so
// we precompute it, bucket batch elements by matrix per step, and run each
// bucket's [<=16 x 64] @ [64 x 64] product on V_WMMA_F32_16X16X4_F32 (the
// exact-fp32 matrix op). connection (4MB) and state (1MB) live in L2.
// Inter-step dependency handled by 511 chained kernel launches on `stream`.
// ---------------------------------------------------------------------------

#define BATCH   4096
#define FDIM    64
#define PATHLEN 512
#define NSTEP   511   // PATHLEN - 1
#define NMAT    256

typedef __attribute__((ext_vector_type(2))) float v2f;
typedef __attribute__((ext_vector_type(8))) float v8f;

// ---------------------------------------------------------------------------
// K0: connT[mat][n][k] = connection[mat][k][n]  (so WMMA B fetches are float2)
// ---------------------------------------------------------------------------
__global__ __launch_bounds__(256) void transpose_conn_kernel(
    const float* __restrict__ conn, float* __restrict__ connT) {
  int mat = blockIdx.x;
  const float* src = conn + mat * (FDIM * FDIM);
  float* dst = connT + mat * (FDIM * FDIM);
  for (int p = threadIdx.x; p < FDIM * FDIM; p += 256) {
    int k = p >> 6;      // row of M (input dim)
    int n = p & 63;      // col of M (output dim)
    dst[n * FDIM + k] = src[p];   // coalesced read, strided write; one-time 4MB
  }
}

// ---------------------------------------------------------------------------
// K1: colmax[t] = max( max_b base_path[b,t], 1.0 )   (exact: max is order-free)
// ---------------------------------------------------------------------------
__global__ __launch_bounds__(256) void colmax_kernel(
    const float* __restrict__ bp, float* __restrict__ colmax) {
  int t = blockIdx.x;
  __shared__ float red[256];
  float m = -__builtin_inff();
  for (int b = threadIdx.x; b < BATCH; b += 256)
    m = fmaxf(m, bp[b * PATHLEN + t]);
  red[threadIdx.x] = m;
  __syncthreads();
  for (int s = 128; s > 0; s >>= 1) {
    if (threadIdx.x < s) red[threadIdx.x] = fmaxf(red[threadIdx.x], red[threadIdx.x + s]);
    __syncthreads();
  }
  if (threadIdx.x == 0) colmax[t] = fmaxf(red[0], 1.0f);
}

// ---------------------------------------------------------------------------
// K2: per-step counting sort of batch elements into 256 matrix buckets.
//     perm[t][pos] = b (u16), boff[t][0..256] = bucket offsets.
//     Within-bucket order is schedule-dependent but numerically irrelevant:
//     each row's product is computed and stored independently.
// ---------------------------------------------------------------------------
__global__ __launch_bounds__(256) void bucket_kernel(
    const float* __restrict__ bp, const float* __restrict__ colmax,
    unsigned short* __restrict__ perm, int* __restrict__ boff) {
  int t = blockIdx.x;
  __shared__ int hist[NMAT];
  __shared__ int offs[NMAT + 1];
  __shared__ int cursor[NMAT];
  float mx = colmax[t];
  for (int i = threadIdx.x; i < NMAT; i += 256) hist[i] = 0;
  __syncthreads();
  for (int b = threadIdx.x; b < BATCH; b += 256) {
    float v = bp[b * PATHLEN + t];
    int idx = (int)(v * 256.0f / mx);          // matches astype(int32): trunc toward 0 (v>=0)
    idx = min(max(idx, 0), NMAT - 1);
    atomicAdd(&hist[idx], 1);
  }
  __syncthreads();
  if (threadIdx.x == 0) {                      // serial scan over 256 bins: negligible
    int acc = 0;
    for (int i = 0; i < NMAT; ++i) { offs[i] = acc; acc += hist[i]; }
    offs[NMAT] = acc;
  }
  __syncthreads();
  for (int i = threadIdx.x; i < NMAT; i += 256) {
    cursor[i] = 0;
    boff[t * (NMAT + 1) + i] = offs[i];
  }
  if (threadIdx.x == 0) boff[t * (NMAT + 1) + NMAT] = offs[NMAT];
  __syncthreads();
  for (int b = threadIdx.x; b < BATCH; b += 256) {
    float v = bp[b * PATHLEN + t];
    int idx = (int)(v * 256.0f / mx);
    idx = min(max(idx, 0), NMAT - 1);
    int pos = offs[idx] + atomicAdd(&cursor[idx], 1);
    perm[t * BATCH + pos] = (unsigned short)b;
  }
}

// ---------------------------------------------------------------------------
// K3 (x511): one block (one wave32) per matrix bucket.
//   chunk of <=16 gathered rows:  OUT[16x64] = V[16x64] @ M[64x64]
//   = 4 N-tiles x 16 K-slices of V_WMMA_F32_16X16X4_F32 (exact fp32).
// Operand layout (ISA 7.12.2, 32-bit, wave32): lane l, m=n=l&15, h=l>>4:
//   A: a = {V[m][4s+2h], V[m][4s+2h+1]}
//   B: b = {M[4s+2h][n], M[4s+2h+1][n]} = float2 load from connT[n][4s+2h]
//   D: vgpr r holds D[r+8h][n]
// ---------------------------------------------------------------------------
__global__ __launch_bounds__(32) void step_kernel(
    const float* __restrict__ src,                 // [BATCH][FDIM]
    float* __restrict__ dst,                       // [BATCH][FDIM]
    const float* __restrict__ connT,               // [NMAT][64(n)][64(k)]
    const unsigned short* __restrict__ perm,       // [BATCH] for this step
    const int* __restrict__ boff) {                // [NMAT+1] for this step
  const int mat    = blockIdx.x;
  const int start  = boff[mat];
  const int end    = boff[mat + 1];
  const int l      = threadIdx.x;
  const int lane16 = l & 15;
  const int hh     = l >> 4;
  const float* MT  = connT + mat * (FDIM * FDIM);

  for (int base = start; base < end; base += 16) {
    int cnt = end - base;
    if (cnt > 16) cnt = 16;

    // A-operand source row for this lane (pad lanes reuse row 0; discarded)
    int rA = (int)perm[base + (lane16 < cnt ? lane16 : 0)];
    const float* vrow = src + rA * FDIM;
    const float* mt0  = MT + (lane16)       * FDIM;   // n-tile 0
    const float* mt1  = MT + (16 + lane16)  * FDIM;   // n-tile 1
    const float* mt2  = MT + (32 + lane16)  * FDIM;   // n-tile 2
    const float* mt3  = MT + (48 + lane16)  * FDIM;   // n-tile 3

    v8f c0 = {}, c1 = {}, c2 = {}, c3 = {};
    #pragma unroll
    for (int s = 0; s < 16; ++s) {
      const int k0 = 4 * s + 2 * hh;
      v2f a  = *(const v2f*)(vrow + k0);
      v2f b0 = *(const v2f*)(mt0 + k0);
      v2f b1 = *(const v2f*)(mt1 + k0);
      v2f b2 = *(const v2f*)(mt2 + k0);
      v2f b3 = *(const v2f*)(mt3 + k0);
      // 8 args: (neg_a, A, neg_b, B, c_mod, C, reuse_a, reuse_b)
      c0 = __builtin_amdgcn_wmma_f32_16x16x4_f32(false, a, false, b0, (short)0, c0, false, false);
      c1 = __builtin_amdgcn_wmma_f32_16x16x4_f32(false, a, false, b1, (short)0, c1, false, false);
      c2 = __builtin_amdgcn_wmma_f32_16x16x4_f32(false, a, false, b2, (short)0, c2, false, false);
      c3 = __builtin_amdgcn_wmma_f32_16x16x4_f32(false, a, false, b3, (short)0, c3, false, false);
    }

    // Scatter valid rows: D row m2 = r + 8*hh, column n = nt*16 + lane16
    #pragma unroll
    for (int r = 0; r < 8; ++r) {
      int m2 = r + 8 * hh;
      if (m2 < cnt) {
        float* drow = dst + (int)perm[base + m2] * FDIM;
        drow[lane16]      = c0[r];
        drow[16 + lane16] = c1[r];
        drow[32 + lane16] = c2[r];
        drow[48 + lane16] = c3[r];
      }
    }
  }
}

// ---------------------------------------------------------------------------
// Host driver. Workspace layout (needs ~11 MB of d_ws):
//   connT   4,194,304 B
//   perm    4,186,112 B (u16[511][4096])
//   boff      525,312 B (int[511][257], padded)
//   colmax      2,048 B
//   buf0    1,048,576 B
//   buf1    1,048,576 B
// ---------------------------------------------------------------------------
extern "C" void kernel_launch(void* const* d_in, const int* in_sizes, int n_in,
                              void* d_out, int out_size, void* d_ws, size_t ws_size,
                              hipStream_t stream) {
  const float* fiber = (const float*)d_in[0];   // [4096][64]
  const float* bp    = (const float*)d_in[1];   // [4096][512]
  const float* conn  = (const float*)d_in[2];   // [256][64][64]
  float* out = (float*)d_out;

  char* ws = (char*)d_ws;
  size_t off = 0;
  float*          connT  = (float*)(ws + off);          off += (size_t)NMAT * FDIM * FDIM * 4;
  unsigned short* perm   = (unsigned short*)(ws + off); off += (size_t)NSTEP * BATCH * 2;
  int*            boff   = (int*)(ws + off);            off += (size_t)NSTEP * (NMAT + 1) * 4 + 4;
  float*          colmax = (float*)(ws + off);          off += (size_t)NSTEP * 4 + 4;
  off = (off + 255) & ~(size_t)255;
  float*          buf0   = (float*)(ws + off);          off += (size_t)BATCH * FDIM * 4;
  float*          buf1   = (float*)(ws + off);          off += (size_t)BATCH * FDIM * 4;
  (void)ws_size; (void)in_sizes; (void)n_in; (void)out_size;

  transpose_conn_kernel<<<NMAT, 256, 0, stream>>>(conn, connT);
  colmax_kernel<<<NSTEP, 256, 0, stream>>>(bp, colmax);
  bucket_kernel<<<NSTEP, 256, 0, stream>>>(bp, colmax, perm, boff);

  for (int t = 0; t < NSTEP; ++t) {
    const float* src = (t == 0) ? fiber : (((t - 1) & 1) ? buf1 : buf0);
    float*       dst = (t == NSTEP - 1) ? out : ((t & 1) ? buf1 : buf0);
    step_kernel<<<NMAT, 32, 0, stream>>>(src, dst, connT,
                                         perm + (size_t)t * BATCH,
                                         boff + (size_t)t * (NMAT + 1));
  }
}